// MyGraphConvModel_11871289606144
// MI455X (gfx1250) — compile-verified
//
#include <hip/hip_runtime.h>
#include <hip/hip_bf16.h>
#include <math.h>

typedef float v2f __attribute__((ext_vector_type(2)));
typedef float v8f __attribute__((ext_vector_type(8)));

#define N_ATOMS 262144
#define BATCH   1024
#define F_IN    75
#define KPAD1   80
#define H1      64
#define H2      128
#define H3      256
#define EPSV    0.001f

__device__ __constant__ int OFFS_c[8] = {0, 8192, 40960, 106496, 172032, 221184, 245760, 262144};

static __device__ __forceinline__ int degree_of(int atom) {
    int d = 0;
#pragma unroll
    for (int i = 1; i <= 6; ++i) d += (atom >= OFFS_c[i]) ? 1 : 0;
    return d;
}

static __device__ __forceinline__ const int* sel_adj(int deg, const int* a1, const int* a2,
                                                     const int* a3, const int* a4,
                                                     const int* a5, const int* a6) {
    switch (deg) {
        case 1: return a1; case 2: return a2; case 3: return a3;
        case 4: return a4; case 5: return a5; default: return a6;
    }
}

static __device__ __forceinline__ v8f wmma_f32x4(v2f a, v2f b, v8f c) {
    return __builtin_amdgcn_wmma_f32_16x16x4_f32(false, a, false, b, (short)0, c, false, false);
}

// ---------------- prep kernels ----------------
__global__ void pad_atoms_k(const float* __restrict__ af, float* __restrict__ ap) {
    int idx = blockIdx.x * blockDim.x + threadIdx.x;            // N_ATOMS*KPAD1 threads
    int atom = idx / KPAD1, c = idx - atom * KPAD1;
    ap[idx] = (c < F_IN) ? af[(size_t)atom * F_IN + c] : 0.f;
}

__global__ void pad_w1_k(const float* __restrict__ w, float* __restrict__ wp) {
    int idx = blockIdx.x * blockDim.x + threadIdx.x;            // 13*KPAD1*H1 threads
    int m = idx / (KPAD1 * H1);
    int r = idx - m * (KPAD1 * H1);
    int k = r / H1, c = r - k * H1;
    wp[idx] = (k < F_IN) ? w[((size_t)m * F_IN + k) * H1 + c] : 0.f;
}

// ---------------- gather / pool (float4 vectorized) ----------------
__global__ void gather_sum4_k(const float* __restrict__ X, float* __restrict__ NB,
                              const int* a1, const int* a2, const int* a3,
                              const int* a4, const int* a5, const int* a6, int C4) {
    int idx = blockIdx.x * blockDim.x + threadIdx.x;
    int atom = idx / C4, c4 = idx - atom * C4;
    int deg = degree_of(atom);
    if (deg == 0) return;                                        // NB unused for deg-0 rows
    const int* adj = sel_adj(deg, a1, a2, a3, a4, a5, a6);
    int lr = atom - OFFS_c[deg];
    const float4* Xv = (const float4*)X;
    float4 s = make_float4(0.f, 0.f, 0.f, 0.f);
    for (int j = 0; j < deg; ++j) {
        float4 v = Xv[(size_t)adj[(size_t)lr * deg + j] * C4 + c4];
        s.x += v.x; s.y += v.y; s.z += v.z; s.w += v.w;
    }
    ((float4*)NB)[(size_t)atom * C4 + c4] = s;
}

__global__ void pool_max4_k(const float* __restrict__ X, float* __restrict__ Y,
                            const int* a1, const int* a2, const int* a3,
                            const int* a4, const int* a5, const int* a6, int C4) {
    int idx = blockIdx.x * blockDim.x + threadIdx.x;
    int atom = idx / C4, c4 = idx - atom * C4;
    int deg = degree_of(atom);
    const float4* Xv = (const float4*)X;
    float4 m = Xv[(size_t)atom * C4 + c4];
    if (deg > 0) {
        const int* adj = sel_adj(deg, a1, a2, a3, a4, a5, a6);
        int lr = atom - OFFS_c[deg];
        for (int j = 0; j < deg; ++j) {
            float4 v = Xv[(size_t)adj[(size_t)lr * deg + j] * C4 + c4];
            m.x = fmaxf(m.x, v.x); m.y = fmaxf(m.y, v.y);
            m.z = fmaxf(m.z, v.z); m.w = fmaxf(m.w, v.w);
        }
    }
    ((float4*)Y)[(size_t)atom * C4 + c4] = m;
}

// ---------------- graph-conv GEMM (WMMA f32 16x16x4), fused bias+tanh+bn ----------------
// block = 128 threads (4 waves), 16 rows; wave w owns col tiles (w + 4t)*16, t<T, T=Cout/64.
// A fragments loaded once per K-step and reused across all T column tiles.
template<int K, int Cout, int T>
__global__ void gemm_conv_k(const float* __restrict__ Xs, const float* __restrict__ Xn,
                            const float* __restrict__ W,   // [13][K][Cout]
                            const float* __restrict__ bias,// [7][Cout]
                            const float* __restrict__ bn,  // [4][Cout]
                            float* __restrict__ Y) {
    const int row0 = blockIdx.x * 16;
    const int wave = threadIdx.x >> 5;
    const int lane = threadIdx.x & 31;
    const int l = lane & 15;
    const int half = lane >> 4;

    const int deg = degree_of(row0);                 // 16-row tiles never cross degree bounds
    const float* Ws; const float* Wn = nullptr; int brow;
    if (deg == 0) { Ws = W + 12 * K * Cout; brow = 6; }
    else { Wn = W + (2 * deg - 2) * K * Cout; Ws = Wn + K * Cout; brow = deg - 1; }

    v8f acc[T];
#pragma unroll
    for (int t = 0; t < T; ++t) acc[t] = v8f{};

    const float* xsp = Xs + (row0 + l) * K + half * 2;
    if (deg == 0) {
#pragma unroll 4
        for (int k = 0; k < K; k += 4) {
            v2f a; a.x = xsp[k]; a.y = xsp[k + 1];
            const float* wrow = Ws + (k + half * 2) * Cout + wave * 16 + l;
#pragma unroll
            for (int t = 0; t < T; ++t) {
                v2f b; b.x = wrow[t * 64]; b.y = wrow[t * 64 + Cout];
                acc[t] = wmma_f32x4(a, b, acc[t]);
            }
        }
    } else {
        const float* xnp = Xn + (row0 + l) * K + half * 2;
#pragma unroll 4
        for (int k = 0; k < K; k += 4) {
            v2f a;  a.x  = xsp[k]; a.y  = xsp[k + 1];
            v2f an; an.x = xnp[k]; an.y = xnp[k + 1];
            const float* wrow_s = Ws + (k + half * 2) * Cout + wave * 16 + l;
            const float* wrow_n = Wn + (k + half * 2) * Cout + wave * 16 + l;
#pragma unroll
            for (int t = 0; t < T; ++t) {
                v2f bq; bq.x = wrow_n[t * 64]; bq.y = wrow_n[t * 64 + Cout];
                acc[t] = wmma_f32x4(an, bq, acc[t]);
                v2f bp; bp.x = wrow_s[t * 64]; bp.y = wrow_s[t * 64 + Cout];
                acc[t] = wmma_f32x4(a, bp, acc[t]);
            }
        }
    }

#pragma unroll
    for (int t = 0; t < T; ++t) {
        const int col = wave * 16 + t * 64 + l;
        const float sc = bn[col] * rsqrtf(bn[3 * Cout + col] + EPSV);
        const float mu = bn[2 * Cout + col];
        const float bb = bn[Cout + col];
        const float bi = bias[brow * Cout + col];
#pragma unroll
        for (int v = 0; v < 8; ++v) {
            int r = row0 + v + 8 * half;
            float y = tanhf(acc[t][v] + bi);
            Y[(size_t)r * Cout + col] = sc * (y - mu) + bb;
        }
    }
}

// ---------------- dense (128->256) + tanh + bn3, writes h3 ----------------
__global__ void gemm_dense_k(const float* __restrict__ X,  // [N][128]
                             const float* __restrict__ W,  // [128][256]
                             const float* __restrict__ bias,// [256]
                             const float* __restrict__ bn, // [4][256]
                             float* __restrict__ H3out) {
    const int row0 = blockIdx.x * 16;
    const int wave = threadIdx.x >> 5;
    const int lane = threadIdx.x & 31;
    const int l = lane & 15;
    const int half = lane >> 4;

    v8f acc[4];
#pragma unroll
    for (int t = 0; t < 4; ++t) acc[t] = v8f{};

    const float* xp = X + (row0 + l) * H2 + half * 2;
#pragma unroll 4
    for (int k = 0; k < H2; k += 4) {
        v2f a; a.x = xp[k]; a.y = xp[k + 1];
        const float* wrow = W + (k + half * 2) * H3 + wave * 16 + l;
#pragma unroll
        for (int t = 0; t < 4; ++t) {
            v2f b; b.x = wrow[t * 64]; b.y = wrow[t * 64 + H3];
            acc[t] = wmma_f32x4(a, b, acc[t]);
        }
    }

#pragma unroll
    for (int t = 0; t < 4; ++t) {
        const int col = wave * 16 + t * 64 + l;
        const float sc = bn[col] * rsqrtf(bn[3 * H3 + col] + EPSV);
        const float mu = bn[2 * H3 + col];
        const float bb = bn[H3 + col];
        const float bi = bias[col];
#pragma unroll
        for (int v = 0; v < 8; ++v) {
            int r = row0 + v + 8 * half;
            float y = tanhf(acc[t][v] + bi);
            H3out[(size_t)r * H3 + col] = sc * (y - mu) + bb;
        }
    }
}

// ---------------- counting-sort inverted index over membership ----------------
__global__ void zero_cnt_k(int* __restrict__ cnt) {
    int i = blockIdx.x * blockDim.x + threadIdx.x;
    if (i < BATCH) cnt[i] = 0;
}
__global__ void hist_k(const int* __restrict__ mem, int* __restrict__ cnt) {
    int i = blockIdx.x * blockDim.x + threadIdx.x;
    atomicAdd(&cnt[mem[i]], 1);
}
__global__ void scan_k(const int* __restrict__ cnt, int* __restrict__ starts,
                       int* __restrict__ cursor) {
    __shared__ int sh[BATCH];
    int t = threadIdx.x;
    int my = cnt[t];
    sh[t] = my;
    __syncthreads();
    for (int off = 1; off < BATCH; off <<= 1) {
        int v = (t >= off) ? sh[t - off] : 0;
        __syncthreads();
        sh[t] += v;
        __syncthreads();
    }
    int excl = sh[t] - my;
    starts[t] = excl;
    cursor[t] = excl;
}
__global__ void scatter_k(const int* __restrict__ mem, int* __restrict__ cursor,
                          int* __restrict__ order) {
    int i = blockIdx.x * blockDim.x + threadIdx.x;
    int pos = atomicAdd(&cursor[mem[i]], 1);
    order[pos] = i;
}

// ---------------- segment sum/max reduce: block per segment, thread per channel ----------------
__global__ void seg_reduce_k(const float* __restrict__ H3v, const int* __restrict__ starts,
                             const int* __restrict__ cnt, const int* __restrict__ order,
                             float* __restrict__ sums, float* __restrict__ maxs) {
    int m = blockIdx.x, c = threadIdx.x;                    // 256 threads
    int s0 = starts[m], n = cnt[m];
    float s = 0.f, mx = -INFINITY;
    int i = 0;
    for (; i + 1 < n; i += 2) {
        int a0 = order[s0 + i], a1 = order[s0 + i + 1];
        float v0 = H3v[(size_t)a0 * H3 + c];
        float v1 = H3v[(size_t)a1 * H3 + c];
        s += v0 + v1;
        mx = fmaxf(mx, fmaxf(v0, v1));
    }
    if (i < n) {
        float v = H3v[(size_t)order[s0 + i] * H3 + c];
        s += v;
        mx = fmaxf(mx, v);
    }
    sums[(size_t)m * H3 + c] = s;
    maxs[(size_t)m * H3 + c] = mx;
}

// ---------------- final: tanh(concat[sum,max]) @ reg_W + reg_b ----------------
__global__ void final_k(const float* __restrict__ sums, const float* __restrict__ maxs,
                        const float* __restrict__ regW, const float* __restrict__ regb,
                        float* __restrict__ out) {
    __shared__ float red[256];
    int m = blockIdx.x, t = threadIdx.x;                     // 256 threads
    float p = tanhf(sums[(size_t)m * H3 + t]) * regW[t];
    p += tanhf(maxs[(size_t)m * H3 + t]) * regW[H3 + t];
    red[t] = p;
    __syncthreads();
    for (int s = 128; s > 0; s >>= 1) {
        if (t < s) red[t] += red[t + s];
        __syncthreads();
    }
    if (t == 0) out[m] = red[0] + regb[0];
}

extern "C" void kernel_launch(void* const* d_in, const int* in_sizes, int n_in,
                              void* d_out, int out_size, void* d_ws, size_t ws_size,
                              hipStream_t stream) {
    const float* af   = (const float*)d_in[0];
    const int*   mem  = (const int*)d_in[1];
    const int *a1 = (const int*)d_in[2], *a2 = (const int*)d_in[3], *a3 = (const int*)d_in[4];
    const int *a4 = (const int*)d_in[5], *a5 = (const int*)d_in[6], *a6 = (const int*)d_in[7];
    const float* gc1W = (const float*)d_in[8];
    const float* gc1b = (const float*)d_in[9];
    const float* bn1  = (const float*)d_in[10];
    const float* gc2W = (const float*)d_in[11];
    const float* gc2b = (const float*)d_in[12];
    const float* bn2  = (const float*)d_in[13];
    const float* dW   = (const float*)d_in[14];
    const float* db   = (const float*)d_in[15];
    const float* bn3  = (const float*)d_in[16];
    const float* regW = (const float*)d_in[17];
    const float* regb = (const float*)d_in[18];
    float* out = (float*)d_out;

    const size_t NA = N_ATOMS;
    float* ws   = (float*)d_ws;
    float* Apad = ws;                         // NA*80  (later h1p NA*64, then h2p NA*128)
    float* NB   = ws + NA * 80;               // NA*80  (NB1; NB2 uses NA*64)
    float* Hbuf = ws + NA * 160;              // NA*256 (h1a NA*64, h2a NA*128, then h3 NA*256)
    float* sums = ws + NA * 416;              // 1024*256
    float* maxs = sums + (size_t)BATCH * H3;  // 1024*256
    float* Wpad = maxs + (size_t)BATCH * H3;  // 13*80*64
    int*   cnt    = (int*)(Wpad + 13 * KPAD1 * H1);
    int*   starts = cnt + BATCH;
    int*   cursor = starts + BATCH;
    int*   order  = cursor + BATCH;           // NA ints

    pad_atoms_k<<<(int)(NA * KPAD1 / 256), 256, 0, stream>>>(af, Apad);
    pad_w1_k<<<(13 * KPAD1 * H1) / 256, 256, 0, stream>>>(gc1W, Wpad);

    // inverted index over membership (independent of the conv pipeline)
    zero_cnt_k<<<BATCH / 256, 256, 0, stream>>>(cnt);
    hist_k<<<(int)(NA / 256), 256, 0, stream>>>(mem, cnt);
    scan_k<<<1, BATCH, 0, stream>>>(cnt, starts, cursor);
    scatter_k<<<(int)(NA / 256), 256, 0, stream>>>(mem, cursor, order);

    // layer 1: gather -> conv(75pad->64) -> pool
    gather_sum4_k<<<(int)(NA * (KPAD1 / 4) / 256), 256, 0, stream>>>(Apad, NB, a1, a2, a3, a4, a5, a6, KPAD1 / 4);
    gemm_conv_k<KPAD1, H1, 1><<<NA / 16, 128, 0, stream>>>(Apad, NB, Wpad, gc1b, bn1, Hbuf);
    pool_max4_k<<<(int)(NA * (H1 / 4) / 256), 256, 0, stream>>>(Hbuf, Apad /*h1p*/, a1, a2, a3, a4, a5, a6, H1 / 4);

    // layer 2: gather -> conv(64->128) -> pool
    gather_sum4_k<<<(int)(NA * (H1 / 4) / 256), 256, 0, stream>>>(Apad, NB, a1, a2, a3, a4, a5, a6, H1 / 4);
    gemm_conv_k<H1, H2, 2><<<NA / 16, 128, 0, stream>>>(Apad, NB, gc2W, gc2b, bn2, Hbuf);
    pool_max4_k<<<(int)(NA * (H2 / 4) / 256), 256, 0, stream>>>(Hbuf, Apad /*h2p*/, a1, a2, a3, a4, a5, a6, H2 / 4);

    // dense(128->256) + tanh + bn3 -> h3
    gemm_dense_k<<<NA / 16, 128, 0, stream>>>(Apad, dW, db, bn3, Hbuf);

    // segment sum/max via inverted index, then tanh(concat) @ reg_W + reg_b
    seg_reduce_k<<<BATCH, 256, 0, stream>>>(Hbuf, starts, cnt, order, sums, maxs);
    final_k<<<BATCH, 256, 0, stream>>>(sums, maxs, regW, regb, out);
}